// TorchNeighborList_24704651887029
// MI455X (gfx1250) — compile-verified
//
#include <hip/hip_runtime.h>
#include <hip/hip_bf16.h>

typedef __attribute__((ext_vector_type(2))) float v2f;
typedef __attribute__((ext_vector_type(8))) float v8f;

#define CUT2 25.0f   // CUTOFF^2

// ---------------------------------------------------------------------------
// prep: sq[j] = |r_j|^2 ; bp[j] = (-2x, -2y, -2z, sq)  (the WMMA B-side rows)
// ---------------------------------------------------------------------------
__global__ void nl_prep(const float* __restrict__ pos, float* __restrict__ sq,
                        float4* __restrict__ bp, int n) {
  int j = blockIdx.x * blockDim.x + threadIdx.x;
  if (j >= n) return;
  float x = pos[3 * j + 0], y = pos[3 * j + 1], z = pos[3 * j + 2];
  float s = x * x + y * y + z * z;
  sq[j] = s;
  bp[j] = make_float4(-2.f * x, -2.f * y, -2.f * z, s);
}

// ---------------------------------------------------------------------------
// Pass 1: per-row neighbor counts. One wave per 16-row strip.
// D tile via V_WMMA_F32_16X16X4_F32:  D = A(16x4) * B(4x16) + C  == d^2 tile.
// A layout (f32 16x4): lanes 0-15 hold K=0(v0),K=1(v1); lanes 16-31 K=2,K=3.
// C/D layout: vgpr v, lanes 0-15 -> row v ; lanes 16-31 -> row v+8 ; col = lane%16.
// ---------------------------------------------------------------------------
__global__ void nl_count(const float* __restrict__ pos,
                         const float* __restrict__ sq,
                         const float4* __restrict__ bp,
                         unsigned* __restrict__ counts, int n) {
  const int lane = threadIdx.x & 31;
  const int wave = threadIdx.x >> 5;
  const int strip = blockIdx.x * (blockDim.x >> 5) + wave;
  const int rowbase = strip * 16;
  if (rowbase >= n) return;                 // wave-uniform

  const int m16 = lane & 15;
  const int half = lane >> 4;

  int ia = rowbase + m16;
  if (ia >= n) ia = n - 1;
  v2f a;
  a.x = half ? pos[3 * ia + 2] : pos[3 * ia + 0];
  a.y = half ? 1.0f            : pos[3 * ia + 1];

  v8f c;
#pragma unroll
  for (int v = 0; v < 8; ++v) {
    int r = rowbase + v + 8 * half;
    c[v] = sq[r < n ? r : (n - 1)];
  }

  unsigned cntL[8], cntH[8];
#pragma unroll
  for (int v = 0; v < 8; ++v) { cntL[v] = 0u; cntH[v] = 0u; }

  const int ntiles = (n + 15) >> 4;
  for (int ct = 0; ct < ntiles; ++ct) {
    int j = ct * 16 + m16;
    int jc = j < n ? j : (n - 1);
    float4 b4 = bp[jc];
    v2f b;
    b.x = half ? b4.z : b4.x;
    b.y = half ? b4.w : b4.y;

    v8f d = __builtin_amdgcn_wmma_f32_16x16x4_f32(
        false, a, false, b, (short)0, c, false, false);

#pragma unroll
    for (int v = 0; v < 8; ++v) {
      int row = rowbase + v + 8 * half;
      bool pred = (d[v] < CUT2) && (j != row) && (j < n) && (row < n);
      unsigned m = (unsigned)__ballot(pred);
      cntL[v] += (unsigned)__popc(m & 0xFFFFu);
      cntH[v] += (unsigned)__popc(m >> 16);
    }
  }

  if (lane == 0) {
#pragma unroll
    for (int v = 0; v < 8; ++v) {
      if (rowbase + v < n)     counts[rowbase + v]     = cntL[v];
      if (rowbase + 8 + v < n) counts[rowbase + 8 + v] = cntH[v];
    }
  }
}

// ---------------------------------------------------------------------------
// Pass 2: single-block exclusive scan of row counts -> offsets[0..n], total at [n]
// ---------------------------------------------------------------------------
__global__ void nl_scan(const unsigned* __restrict__ counts,
                        unsigned* __restrict__ offsets, int n) {
  __shared__ unsigned tmp[1024];
  __shared__ unsigned carry;
  if (threadIdx.x == 0) carry = 0u;
  __syncthreads();
  for (int base = 0; base < n; base += 1024) {
    int idx = base + (int)threadIdx.x;
    unsigned v = (idx < n) ? counts[idx] : 0u;
    tmp[threadIdx.x] = v;
    __syncthreads();
    for (int d = 1; d < 1024; d <<= 1) {
      unsigned t = (threadIdx.x >= (unsigned)d) ? tmp[threadIdx.x - d] : 0u;
      __syncthreads();
      tmp[threadIdx.x] += t;
      __syncthreads();
    }
    if (idx < n) offsets[idx] = carry + (tmp[threadIdx.x] - v);
    __syncthreads();
    if (threadIdx.x == 0) carry += tmp[1023];
    __syncthreads();
  }
  if (threadIdx.x == 0) offsets[n] = carry;
}

// ---------------------------------------------------------------------------
// zero the whole output (padding entries must be pairs=(0,0), scale=0, ds=0)
// ---------------------------------------------------------------------------
__global__ void nl_zero(int* __restrict__ out, int nelem) {
  int i = blockIdx.x * blockDim.x + threadIdx.x;
  int stride = gridDim.x * blockDim.x;
  for (; i < nelem; i += stride) out[i] = 0;
}

// ---------------------------------------------------------------------------
// Pass 3: recompute tiles (WMMA) in ascending-j order, compacted writes.
// Per hit: pairs[k]={i,j}, scale=(i<j), ds = exact |r_j - r_i|.
// ---------------------------------------------------------------------------
__global__ void nl_write(const float* __restrict__ pos,
                         const float* __restrict__ sq,
                         const float4* __restrict__ bp,
                         const unsigned* __restrict__ offsets,
                         int n, int maxPairs,
                         int* __restrict__ pairs, int* __restrict__ scales,
                         float* __restrict__ ds) {
  const int lane = threadIdx.x & 31;
  const int wave = threadIdx.x >> 5;
  const int strip = blockIdx.x * (blockDim.x >> 5) + wave;
  const int rowbase = strip * 16;
  if (rowbase >= n) return;

  const int m16 = lane & 15;
  const int half = lane >> 4;

  int ia = rowbase + m16;
  if (ia >= n) ia = n - 1;
  v2f a;
  a.x = half ? pos[3 * ia + 2] : pos[3 * ia + 0];
  a.y = half ? 1.0f            : pos[3 * ia + 1];

  v8f c;
#pragma unroll
  for (int v = 0; v < 8; ++v) {
    int r = rowbase + v + 8 * half;
    c[v] = sq[r < n ? r : (n - 1)];
  }

  unsigned offL[8], offH[8];
#pragma unroll
  for (int v = 0; v < 8; ++v) {
    int rl = rowbase + v, rh = rowbase + 8 + v;
    offL[v] = (rl < n) ? offsets[rl] : 0u;
    offH[v] = (rh < n) ? offsets[rh] : 0u;
  }

  const unsigned below_me = (1u << lane) - 1u;
  const int ntiles = (n + 15) >> 4;
  for (int ct = 0; ct < ntiles; ++ct) {
    int j = ct * 16 + m16;
    int jc = j < n ? j : (n - 1);
    float4 b4 = bp[jc];
    v2f b;
    b.x = half ? b4.z : b4.x;
    b.y = half ? b4.w : b4.y;

    v8f d = __builtin_amdgcn_wmma_f32_16x16x4_f32(
        false, a, false, b, (short)0, c, false, false);

#pragma unroll
    for (int v = 0; v < 8; ++v) {
      int row = rowbase + v + 8 * half;
      bool pred = (d[v] < CUT2) && (j != row) && (j < n) && (row < n);
      unsigned m = (unsigned)__ballot(pred);
      unsigned below = m & below_me;
      if (half) below &= 0xFFFF0000u;        // only this row's (high) lanes
      unsigned myoff = (half ? offH[v] : offL[v]) + (unsigned)__popc(below);
      if (pred && myoff < (unsigned)maxPairs) {
        pairs[2 * myoff + 0] = row;
        pairs[2 * myoff + 1] = j;
        scales[myoff] = (row < j) ? 1 : 0;
        float dx = pos[3 * j + 0] - pos[3 * row + 0];
        float dy = pos[3 * j + 1] - pos[3 * row + 1];
        float dz = pos[3 * j + 2] - pos[3 * row + 2];
        ds[myoff] = sqrtf(dx * dx + dy * dy + dz * dz);
      }
      offL[v] += (unsigned)__popc(m & 0xFFFFu);
      offH[v] += (unsigned)__popc(m >> 16);
    }
  }
}

// ---------------------------------------------------------------------------
extern "C" void kernel_launch(void* const* d_in, const int* in_sizes, int n_in,
                              void* d_out, int out_size, void* d_ws, size_t ws_size,
                              hipStream_t stream) {
  const float* pos = (const float*)d_in[0];
  const int n = in_sizes[0] / 3;            // 8000
  const int maxPairs = out_size / 4;        // pairs(2M) + scales(M) + ds(M)

  // workspace layout (16B aligned slices)
  char* ws = (char*)d_ws;
  size_t off = 0;
  float* sq = (float*)(ws + off);           off += ((size_t)n * 4 + 15) & ~15ull;
  float4* bp = (float4*)(ws + off);         off += (size_t)n * 16;
  unsigned* counts = (unsigned*)(ws + off); off += ((size_t)n * 4 + 15) & ~15ull;
  unsigned* offsets = (unsigned*)(ws + off);

  int* outi = (int*)d_out;
  int* pairs = outi;                        // [maxPairs][2]
  int* scales = outi + 2 * (size_t)maxPairs;
  float* dsv = (float*)(outi + 3 * (size_t)maxPairs);

  const int strips = (n + 15) / 16;
  const int wavesPerBlock = 8;              // 256 threads = 8 wave32
  const int stripBlocks = (strips + wavesPerBlock - 1) / wavesPerBlock;

  nl_prep<<<(n + 255) / 256, 256, 0, stream>>>(pos, sq, bp, n);
  nl_count<<<stripBlocks, 256, 0, stream>>>(pos, sq, bp, counts, n);
  nl_scan<<<1, 1024, 0, stream>>>(counts, offsets, n);
  {
    int zblocks = (out_size + 255) / 256;
    if (zblocks > 2048) zblocks = 2048;
    nl_zero<<<zblocks, 256, 0, stream>>>(outi, out_size);
  }
  nl_write<<<stripBlocks, 256, 0, stream>>>(pos, sq, bp, offsets, n, maxPairs,
                                            pairs, scales, dsv);
}